// TELightGCN_55061480734869
// MI455X (gfx1250) — compile-verified
//
#include <hip/hip_runtime.h>

#define N_USERS  200000
#define N_ITEMS  100000
#define N_TOPICS 100
#define N_NODES  (N_USERS + N_ITEMS + N_TOPICS)   // 300100
#define N_EDGES  5000000
#define DIM      64
#define LAYERS   3
#define BATCH    16384

typedef __attribute__((ext_vector_type(2))) float v2f;
typedef __attribute__((ext_vector_type(8))) float v8f;

// ---------------------------------------------------------------------------
// h0 = concat(user_emb, item_emb, topic_emb)   (N_NODES x 64, float4 chunks)
// ---------------------------------------------------------------------------
__global__ __launch_bounds__(256) void lgcn_concat(
    const float* __restrict__ ue, const float* __restrict__ ie,
    const float* __restrict__ te, float* __restrict__ h)
{
  unsigned tid = blockIdx.x * blockDim.x + threadIdx.x;
  if (tid >= (unsigned)N_NODES * 16u) return;
  unsigned node = tid >> 4, c = tid & 15u;
  const float* src;
  if (node < (unsigned)N_USERS)                     src = ue + (size_t)node * DIM;
  else if (node < (unsigned)(N_USERS + N_ITEMS))    src = ie + (size_t)(node - N_USERS) * DIM;
  else                                              src = te + (size_t)(node - N_USERS - N_ITEMS) * DIM;
  reinterpret_cast<float4*>(h)[tid] = reinterpret_cast<const float4*>(src)[c];
}

// ---------------------------------------------------------------------------
// zero a float4 array
// ---------------------------------------------------------------------------
__global__ __launch_bounds__(256) void lgcn_zero(float4* __restrict__ p, unsigned n4)
{
  unsigned tid = blockIdx.x * blockDim.x + threadIdx.x;
  if (tid < n4) p[tid] = make_float4(0.f, 0.f, 0.f, 0.f);
}

// ---------------------------------------------------------------------------
// hD[rows[e]] += vals[e] * hS[cols[e]]   — 16 lanes per edge, float4 per lane.
// 16 consecutive lanes touch one 256B-contiguous row (perfectly coalesced);
// scatter uses hardware global_atomic_add_f32 (L2 atomic units; table fits L2).
// ---------------------------------------------------------------------------
__global__ __launch_bounds__(256) void lgcn_spmm(
    const float* __restrict__ hS, float* __restrict__ hD,
    const int* __restrict__ rows, const int* __restrict__ cols,
    const float* __restrict__ vals)
{
  unsigned tid = blockIdx.x * blockDim.x + threadIdx.x;
  if (tid >= (unsigned)N_EDGES * 16u) return;           // 80M threads, fits u32
  unsigned e = tid >> 4, c = tid & 15u;
  int   col = cols[e];
  int   row = rows[e];
  float w   = vals[e];

  // Pre-touch a gather row ~512 edges ahead (lowers gather latency; emits
  // global_prefetch_b8 on gfx1250).
  if (c == 0u) {
    unsigned ep = e + 512u; if (ep >= (unsigned)N_EDGES) ep = e;
    __builtin_prefetch(hS + (size_t)cols[ep] * DIM, 0, 1);
  }

  float4 s = reinterpret_cast<const float4*>(hS + (size_t)col * DIM)[c];
  float* d = hD + (size_t)row * DIM + (size_t)c * 4u;
  unsafeAtomicAdd(d + 0, s.x * w);
  unsafeAtomicAdd(d + 1, s.y * w);
  unsafeAtomicAdd(d + 2, s.z * w);
  unsafeAtomicAdd(d + 3, s.w * w);
}

// ---------------------------------------------------------------------------
// dst[b] (+)= h[rowOff + idx[b]]   (float4 per lane, 16 lanes per row)
// ---------------------------------------------------------------------------
template <bool ACC>
__global__ __launch_bounds__(256) void lgcn_gather(
    float* __restrict__ dst, const float* __restrict__ h,
    const int* __restrict__ idx, int rowOff)
{
  unsigned tid = blockIdx.x * blockDim.x + threadIdx.x;
  if (tid >= (unsigned)BATCH * 16u) return;
  unsigned b = tid >> 4, c = tid & 15u;
  size_t row = (size_t)(rowOff + idx[b]);
  float4 v = reinterpret_cast<const float4*>(h + row * DIM)[c];
  float4* dp = reinterpret_cast<float4*>(dst) + tid;
  if (ACC) {
    float4 o = *dp;
    o.x += v.x; o.y += v.y; o.z += v.z; o.w += v.w;
    *dp = o;
  } else {
    *dp = v;
  }
}

// ---------------------------------------------------------------------------
// out[b] = (1/16) * dot(U[b,:], I[b,:]) via V_WMMA_F32_16X16X4_F32.
// One wave32 per 16 batch rows: C(16x16) = U_tile(16x64) @ I_tileT(64x16),
// chained over 16 K=4 steps; diagonal extracted per the §7.12.2 C layout
// (diag m<8 -> lane m, VGPR m; diag m>=8 -> lane 16+m, VGPR m-8).
// A layout (16x4 f32): lanes 0-15 hold K=0,1; lanes 16-31 hold K=2,3 — the
// per-lane address is row = base + (lane&15), koff = 2*(lane>>4); B (=I^T)
// uses the mirrored K striping, giving the identical addressing pattern.
// ---------------------------------------------------------------------------
__global__ __launch_bounds__(256) void lgcn_dot_wmma(
    const float* __restrict__ U, const float* __restrict__ I,
    float* __restrict__ out)
{
  unsigned lane = threadIdx.x & 31u;
  unsigned wave = (blockIdx.x * blockDim.x + threadIdx.x) >> 5;
  unsigned base = wave * 16u;
  unsigned row  = base + (lane & 15u);
  unsigned koff = (lane >> 4) * 2u;
  const float* up = U + (size_t)row * DIM + koff;
  const float* ip = I + (size_t)row * DIM + koff;

  v8f c = {0.f, 0.f, 0.f, 0.f, 0.f, 0.f, 0.f, 0.f};
#pragma unroll
  for (int k = 0; k < DIM / 4; ++k) {
    v2f a = *reinterpret_cast<const v2f*>(up + 4 * k);
    v2f b = *reinterpret_cast<const v2f*>(ip + 4 * k);
    // 8 args: (neg_a, A, neg_b, B, c_mod, C, reuse_a, reuse_b)
    c = __builtin_amdgcn_wmma_f32_16x16x4_f32(
        false, a, false, b, (short)0, c, false, false);
  }

  float dv = 0.f;
#pragma unroll
  for (int i = 0; i < 8; ++i) {
    dv = (lane == (unsigned)i)        ? c[i] : dv;   // diag m = i
    dv = (lane == (unsigned)(24 + i)) ? c[i] : dv;   // diag m = 8 + i
  }
  // out = (acc/4)·(acc/4) = (1/16) * u_acc·it_acc
  if (lane < 8u)         out[base + lane]         = dv * 0.0625f;
  else if (lane >= 24u)  out[base + (lane - 16u)] = dv * 0.0625f;
}

// ---------------------------------------------------------------------------
extern "C" void kernel_launch(void* const* d_in, const int* in_sizes, int n_in,
                              void* d_out, int out_size, void* d_ws, size_t ws_size,
                              hipStream_t stream)
{
  (void)in_sizes; (void)n_in; (void)out_size; (void)ws_size;

  const int*   users = (const int*)  d_in[0];
  const int*   items = (const int*)  d_in[1];
  const int*   erow  = (const int*)  d_in[2];
  const int*   ecol  = (const int*)  d_in[3];
  const float* eval_ = (const float*)d_in[4];
  const float* ue    = (const float*)d_in[5];
  const float* ie    = (const float*)d_in[6];
  const float* te    = (const float*)d_in[7];
  float* out = (float*)d_out;

  // Workspace: hA | hB (ping-pong node tables) | u_acc | it_acc
  const size_t hfloats = (size_t)N_NODES * DIM;
  float* hA = (float*)d_ws;
  float* hB = hA + hfloats;
  float* ua = hB + hfloats;
  float* ia = ua + (size_t)BATCH * DIM;

  auto nblk = [](size_t n, int b) { return (unsigned)((n + (size_t)b - 1) / (size_t)b); };

  // Layer 0: build h0, seed accumulators with gathered rows.
  lgcn_concat<<<nblk((size_t)N_NODES * 16, 256), 256, 0, stream>>>(ue, ie, te, hA);
  lgcn_gather<false><<<nblk((size_t)BATCH * 16, 256), 256, 0, stream>>>(ua, hA, users, 0);
  lgcn_gather<false><<<nblk((size_t)BATCH * 16, 256), 256, 0, stream>>>(ia, hA, items, N_USERS);

  float* src = hA;
  float* dst = hB;
  for (int l = 0; l < LAYERS; ++l) {
    lgcn_zero<<<nblk((size_t)N_NODES * 16, 256), 256, 0, stream>>>(
        (float4*)dst, (unsigned)((size_t)N_NODES * 16));
    lgcn_spmm<<<nblk((size_t)N_EDGES * 16, 256), 256, 0, stream>>>(
        src, dst, erow, ecol, eval_);
    lgcn_gather<true><<<nblk((size_t)BATCH * 16, 256), 256, 0, stream>>>(ua, dst, users, 0);
    lgcn_gather<true><<<nblk((size_t)BATCH * 16, 256), 256, 0, stream>>>(ia, dst, items, N_USERS);
    float* t = src; src = dst; dst = t;
  }

  // 1024 waves, 16 batch rows each -> 128 blocks of 256 threads.
  lgcn_dot_wmma<<<(BATCH / 16) * 32 / 256, 256, 0, stream>>>(ua, ia, out);
}